// GraphSAGE_11106785427687
// MI455X (gfx1250) — compile-verified
//
#include <hip/hip_runtime.h>
#include <hip/hip_bf16.h>

#define NN   100000
#define EE   3200000
#define DD   128

typedef float v2f __attribute__((ext_vector_type(2)));
typedef float v8f __attribute__((ext_vector_type(8)));

// ---------------------------------------------------------------------------
// zero fill (float4 granularity; all sizes used are multiples of 4 floats)
// ---------------------------------------------------------------------------
__global__ __launch_bounds__(256) void k_zero(float4* p, long n4) {
    long i = (long)blockIdx.x * blockDim.x + threadIdx.x;
    if (i < n4) p[i] = make_float4(0.f, 0.f, 0.f, 0.f);
}

// ---------------------------------------------------------------------------
// degree: deg[dst[e]] += 1.0f   (hardware global_atomic_add_f32)
// ---------------------------------------------------------------------------
__global__ __launch_bounds__(256) void k_degree(const int* __restrict__ dst,
                                                float* __restrict__ deg, int E) {
    long e = (long)blockIdx.x * blockDim.x + threadIdx.x;
    if (e < E) unsafeAtomicAdd(deg + dst[e], 1.0f);
}

// deg -> deg_inv in place (counts are exact integers in f32)
__global__ __launch_bounds__(256) void k_deginv(float* __restrict__ deg, int n) {
    long i = (long)blockIdx.x * blockDim.x + threadIdx.x;
    if (i < n) {
        float d = deg[i];
        deg[i] = (d > 0.f) ? (1.0f / d) : 0.0f;
    }
}

// ---------------------------------------------------------------------------
// edge scatter: one wave per edge, 4 floats (float4) per lane.
// Gather x[src] is a fully-coalesced 512B wave read; scatter is 4 hardware
// f32 atomics per lane into agg[dst].
// ---------------------------------------------------------------------------
__global__ __launch_bounds__(256) void k_scatter(const float* __restrict__ x,
                                                 const int* __restrict__ src,
                                                 const int* __restrict__ dst,
                                                 float* __restrict__ agg, int E) {
    long gid  = (long)blockIdx.x * blockDim.x + threadIdx.x;
    long edge = gid >> 5;            // 32 lanes per edge
    int  c    = (int)(gid & 31);     // float4 chunk 0..31 (covers 128 floats)
    if (edge >= E) return;
    int s = src[edge];
    int d = dst[edge];
    float4 v = ((const float4*)(x + (long)s * DD))[c];
    float* base = agg + (long)d * DD + c * 4;
    unsafeAtomicAdd(base + 0, v.x);
    unsafeAtomicAdd(base + 1, v.y);
    unsafeAtomicAdd(base + 2, v.z);
    unsafeAtomicAdd(base + 3, v.w);
}

// ---------------------------------------------------------------------------
// Fused SAGE linear: out = (agg*deg_inv) @ Wl^T + b + x @ Wr^T  [+ ReLU]
// Block = 256 threads = 8 waves. Block handles 64 rows; wave w handles output
// column tile n0 = 16*w across 4 row tiles. Full-precision f32 WMMA 16x16x4.
// ---------------------------------------------------------------------------
template <bool RELU>
__global__ __launch_bounds__(256) void k_sage_gemm(const float* __restrict__ agg,
                                                   const float* __restrict__ deginv,
                                                   const float* __restrict__ x,
                                                   const float* __restrict__ Wl,
                                                   const float* __restrict__ bl,
                                                   const float* __restrict__ Wr,
                                                   float* __restrict__ out) {
    __shared__ float sAgg[64 * DD];   // 32 KB, deg_inv pre-applied
    __shared__ float sX[64 * DD];     // 32 KB

    const long rowBase = (long)blockIdx.x * 64;

    // ---- stage 64x128 tiles of agg (scaled) and x into LDS, float4 loads ----
    for (int i = threadIdx.x; i < 64 * (DD / 4); i += 256) {
        int  r  = i >> 5;            // row in tile
        int  c4 = i & 31;            // float4 column
        long gr = rowBase + r;
        float4 va = make_float4(0.f, 0.f, 0.f, 0.f);
        float4 vx = make_float4(0.f, 0.f, 0.f, 0.f);
        float  sc = 0.f;
        if (gr < NN) {
            sc = deginv[gr];
            va = ((const float4*)(agg + gr * DD))[c4];
            vx = ((const float4*)(x   + gr * DD))[c4];
        }
        va.x *= sc; va.y *= sc; va.z *= sc; va.w *= sc;
        ((float4*)sAgg)[i] = va;
        ((float4*)sX)[i]   = vx;
    }
    __syncthreads();

    const int wave = threadIdx.x >> 5;   // 0..7 -> N tile
    const int lane = threadIdx.x & 31;
    const int n0   = wave * 16;          // output column base
    const int lm   = lane & 15;
    const int hi   = lane >> 4;          // 0: K=k0,k0+1 ; 1: K=k0+2,k0+3

    // C/D layout: vgpr r -> M = r + 8*hi, N = n0 + lm. Bias depends on N only.
    v8f acc[4];
    {
        float b = bl[n0 + lm];
        for (int mt = 0; mt < 4; ++mt)
            for (int r = 0; r < 8; ++r) acc[mt][r] = b;
    }

    // ---- two GEMM passes: (agg*deg_inv)@Wl^T then x@Wr^T, K in steps of 4 ----
    for (int pass = 0; pass < 2; ++pass) {
        const float* __restrict__ sA = pass ? sX : sAgg;
        const float* __restrict__ W  = pass ? Wr : Wl;
        // B[k][n] = W[n][k]; lane reads W[n0+lm][kk], W[n0+lm][kk+1]
        const float* wrow = W + (long)(n0 + lm) * DD;
        for (int k0 = 0; k0 < DD; k0 += 4) {
            const int kk = k0 + 2 * hi;
            v2f bfrag = *(const v2f*)(wrow + kk);           // global (L2-hot)
            for (int mt = 0; mt < 4; ++mt) {
                v2f afrag = *(const v2f*)(sA + (mt * 16 + lm) * DD + kk); // LDS
                acc[mt] = __builtin_amdgcn_wmma_f32_16x16x4_f32(
                    false, afrag, false, bfrag, (short)0, acc[mt], false, false);
            }
        }
    }

    // ---- epilogue: optional ReLU, store ----
    for (int mt = 0; mt < 4; ++mt) {
        for (int r = 0; r < 8; ++r) {
            long row = rowBase + mt * 16 + r + hi * 8;
            if (row < NN) {
                float v = acc[mt][r];
                if (RELU) v = fmaxf(v, 0.f);
                out[row * DD + n0 + lm] = v;
            }
        }
    }
}

// ---------------------------------------------------------------------------
// row-wise L2 normalize in place: one wave per row, float4/lane, shuffle sum
// ---------------------------------------------------------------------------
__global__ __launch_bounds__(256) void k_normalize(float* __restrict__ p, int n) {
    int wave = threadIdx.x >> 5;
    int lane = threadIdx.x & 31;
    long row = (long)blockIdx.x * 8 + wave;
    if (row >= n) return;
    float4* rp = (float4*)(p + row * DD);
    float4 v = rp[lane];
    float ss = v.x * v.x + v.y * v.y + v.z * v.z + v.w * v.w;
    #pragma unroll
    for (int off = 16; off > 0; off >>= 1) ss += __shfl_xor(ss, off, 32);
    float nrm = sqrtf(ss);
    float inv = 1.0f / fmaxf(nrm, 1e-12f);
    v.x *= inv; v.y *= inv; v.z *= inv; v.w *= inv;
    rp[lane] = v;
}

// ---------------------------------------------------------------------------
extern "C" void kernel_launch(void* const* d_in, const int* in_sizes, int n_in,
                              void* d_out, int out_size, void* d_ws, size_t ws_size,
                              hipStream_t stream) {
    const int*   edge = (const int*)d_in[0];      // [2, E] int32
    const float* emb  = (const float*)d_in[1];    // [N, D]
    const float* Wl0  = (const float*)d_in[2];
    const float* b0   = (const float*)d_in[3];
    const float* Wr0  = (const float*)d_in[4];
    const float* Wl1  = (const float*)d_in[5];
    const float* b1   = (const float*)d_in[6];
    const float* Wr1  = (const float*)d_in[7];
    float* out = (float*)d_out;

    const int* src = edge;
    const int* dst = edge + EE;

    // workspace carve-up (512B aligned)
    char*  ws  = (char*)d_ws;
    float* deg = (float*)ws;                                   // N floats
    size_t off = ((size_t)NN * 4 + 511) & ~(size_t)511;
    float* agg = (float*)(ws + off);                           // N*D floats
    off += ((size_t)NN * DD * 4 + 511) & ~(size_t)511;
    float* h1  = (float*)(ws + off);                           // N*D floats

    const long aggN4 = (long)NN * DD / 4;
    const int  ZB_AGG = (int)((aggN4 + 255) / 256);
    const int  ZB_DEG = (NN / 4 + 255) / 256;
    const int  EB     = (EE + 255) / 256;
    const int  SB     = (int)(((long)EE * 32 + 255) / 256);
    const int  NB     = (NN + 255) / 256;
    const int  GB     = (NN + 63) / 64;
    const int  NRB    = (NN + 7) / 8;

    // degrees + reciprocal
    k_zero<<<ZB_DEG, 256, 0, stream>>>((float4*)deg, NN / 4);
    k_degree<<<EB, 256, 0, stream>>>(dst, deg, EE);
    k_deginv<<<NB, 256, 0, stream>>>(deg, NN);

    // ---- layer 0 ----
    k_zero<<<ZB_AGG, 256, 0, stream>>>((float4*)agg, aggN4);
    k_scatter<<<SB, 256, 0, stream>>>(emb, src, dst, agg, EE);
    k_sage_gemm<true><<<GB, 256, 0, stream>>>(agg, deg, emb, Wl0, b0, Wr0, h1);

    // ---- layer 1 ----
    k_zero<<<ZB_AGG, 256, 0, stream>>>((float4*)agg, aggN4);
    k_scatter<<<SB, 256, 0, stream>>>(h1, src, dst, agg, EE);
    k_sage_gemm<false><<<GB, 256, 0, stream>>>(agg, deg, h1, Wl1, b1, Wr1, out);

    // ---- L2 normalize ----
    k_normalize<<<NRB, 256, 0, stream>>>(out, NN);
}